// O2U_8280696947088
// MI455X (gfx1250) — compile-verified
//
#include <hip/hip_runtime.h>

// ---------------------------------------------------------------------------
// NGCF-style 3-layer graph conv for MI455X (gfx1250, wave32, WMMA).
//
// Roofline reasoning:
//  - SpMM (6.4M nnz x 128 f32): bandwidth/atomic bound. ego (102MB) + side
//    (102MB) ~ fit 192MB L2 -> gather ego rows (float4/lane) and scatter with
//    no-return global_atomic_add_f32 into side.
//  - Dense: 2 x [200000x128x128] fp32 GEMMs/layer -> V_WMMA_F32_16X16X4_F32,
//    both GEMMs chained through one f32 accumulator (D = S@W1^T + (E.S)@W2^T),
//    epilogue (bias + leaky_relu + L2 row-norm + residual add) fused so the
//    activations make exactly one HBM/L2 round trip per layer.
// ---------------------------------------------------------------------------

#define N_USERS   100000
#define N_OUTFITS 100000
#define N_NODES   (N_USERS + N_OUTFITS)   // 200000
#define EMBED     128
#define N_LAYERS  3
#define NNZ_E     6400000

typedef float v2f __attribute__((ext_vector_type(2)));
typedef float v8f __attribute__((ext_vector_type(8)));

// ego = concat(user_emb[u_id], o_embedding); out = ego (layer-0 term)
__global__ void o2u_init_kernel(const float* __restrict__ user_emb,
                                const int*   __restrict__ u_id,
                                const float* __restrict__ o_emb,
                                float* __restrict__ ego,
                                float* __restrict__ out) {
  size_t tid  = (size_t)blockIdx.x * blockDim.x + threadIdx.x; // float4 units
  size_t node = tid >> 5;                                      // 32 float4/row
  int    q    = (int)(tid & 31);
  if (node >= N_NODES) return;
  const float4* src;
  if (node < N_USERS)
    src = (const float4*)(user_emb + (size_t)u_id[node] * EMBED);
  else
    src = (const float4*)(o_emb + (node - (size_t)N_USERS) * EMBED);
  float4 v = src[q];
  ((float4*)ego)[tid] = v;
  ((float4*)out)[tid] = v;
}

__global__ void o2u_zero_kernel(float4* __restrict__ p, long n4) {
  long tid = (long)blockIdx.x * blockDim.x + threadIdx.x;
  if (tid < n4) p[tid] = make_float4(0.f, 0.f, 0.f, 0.f);
}

// One wave per edge: 32 lanes x float4 gather of ego[col], scaled atomic
// scatter into side[row]. Relaxed agent-scope f32 adds -> global_atomic_add_f32.
__global__ void o2u_spmm_kernel(const int*   __restrict__ erow,
                                const int*   __restrict__ ecol,
                                const float* __restrict__ eval,
                                const float* __restrict__ ego,
                                float* __restrict__ side) {
  int wave = blockIdx.x * (blockDim.x >> 5) + (threadIdx.x >> 5);
  int lane = threadIdx.x & 31;
  if (wave >= NNZ_E) return;
  int   r = erow[wave];
  int   c = ecol[wave];
  float v = eval[wave];
  float4 e = ((const float4*)(ego + (size_t)c * EMBED))[lane];
  float* dst = side + (size_t)r * EMBED + lane * 4;
  __hip_atomic_fetch_add(dst + 0, v * e.x, __ATOMIC_RELAXED, __HIP_MEMORY_SCOPE_AGENT);
  __hip_atomic_fetch_add(dst + 1, v * e.y, __ATOMIC_RELAXED, __HIP_MEMORY_SCOPE_AGENT);
  __hip_atomic_fetch_add(dst + 2, v * e.z, __ATOMIC_RELAXED, __HIP_MEMORY_SCOPE_AGENT);
  __hip_atomic_fetch_add(dst + 3, v * e.w, __ATOMIC_RELAXED, __HIP_MEMORY_SCOPE_AGENT);
}

// Fused layer: D = side @ W1^T + (ego .* side) @ W2^T + (b1+b2),
// ego_new = leaky_relu(D); out += ego_new / max(||ego_new||_2, eps);
// ego <- ego_new (in place, waves own disjoint 16-row strips).
// One wave = 16 rows x 128 cols = 8 C-tiles of 16x16, K marched in steps of 4
// with V_WMMA_F32_16X16X4_F32. Weights transposed in LDS for conflict-free
// B-fragment reads.
__global__ __launch_bounds__(256) void o2u_layer_kernel(
    const float* __restrict__ side,
    float* __restrict__ ego,
    const float* __restrict__ w1, const float* __restrict__ b1,
    const float* __restrict__ w2, const float* __restrict__ b2,
    float* __restrict__ out) {
  __shared__ float w1t[EMBED * EMBED];   // w1t[k*128+j] = W1[j][k]
  __shared__ float w2t[EMBED * EMBED];
  __shared__ float bsum[EMBED];

  const int tid = threadIdx.x;
  for (int t = tid; t < EMBED * EMBED; t += 256) {
    int j = t >> 7, k = t & 127;           // coalesced global read
    w1t[k * EMBED + j] = w1[t];
    w2t[k * EMBED + j] = w2[t];
  }
  if (tid < EMBED) bsum[tid] = b1[tid] + b2[tid];
  __syncthreads();

  const int wid  = tid >> 5;
  const int lane = tid & 31;
  const int row0 = blockIdx.x * 128 + wid * 16;  // N_NODES % 16 == 0
  if (row0 >= N_NODES) return;

  const int h = lane >> 4;    // half-wave: K sub-pair for A/B, M+8 for C/D
  const int m = lane & 15;    // A row within tile; B/C column within tile

  const float* sA = side + (size_t)(row0 + m) * EMBED + 2 * h;
  const float* eA = ego  + (size_t)(row0 + m) * EMBED + 2 * h;

  v8f acc[8];
#pragma unroll
  for (int j = 0; j < 8; ++j) acc[j] = {};

  for (int k = 0; k < EMBED; k += 4) {
    float2 s2 = *(const float2*)(sA + k);
    float2 e2 = *(const float2*)(eA + k);
    v2f a_s; a_s.x = s2.x;        a_s.y = s2.y;          // side fragment
    v2f a_b; a_b.x = s2.x * e2.x; a_b.y = s2.y * e2.y;   // ego.*side fragment
    const int kb0 = (k + 2 * h) * EMBED;
    const int kb1 = kb0 + EMBED;
#pragma unroll
    for (int j = 0; j < 8; ++j) {
      const int cidx = j * 16 + m;
      v2f bw1; bw1.x = w1t[kb0 + cidx]; bw1.y = w1t[kb1 + cidx];
      v2f bw2; bw2.x = w2t[kb0 + cidx]; bw2.y = w2t[kb1 + cidx];
      acc[j] = __builtin_amdgcn_wmma_f32_16x16x4_f32(
          false, a_s, false, bw1, (short)0, acc[j], false, false);
      acc[j] = __builtin_amdgcn_wmma_f32_16x16x4_f32(
          false, a_b, false, bw2, (short)0, acc[j], false, false);
    }
  }

  // Epilogue: bias + leaky relu, per-row squared sums (C layout: VGPR v holds
  // row v (lanes 0-15) / v+8 (lanes 16-31), col = 16j + m).
  float ss[8];
#pragma unroll
  for (int v = 0; v < 8; ++v) ss[v] = 0.f;
#pragma unroll
  for (int j = 0; j < 8; ++j) {
    float bias = bsum[j * 16 + m];
#pragma unroll
    for (int v = 0; v < 8; ++v) {
      float x = acc[j][v] + bias;
      x = x > 0.f ? x : 0.01f * x;
      acc[j][v] = x;
      ss[v] += x * x;
    }
  }
  // Row reduction across the 16 lanes of each half-wave (xor<=8 stays in half).
#pragma unroll
  for (int v = 0; v < 8; ++v) {
    float s = ss[v];
    s += __shfl_xor(s, 1, 32);
    s += __shfl_xor(s, 2, 32);
    s += __shfl_xor(s, 4, 32);
    s += __shfl_xor(s, 8, 32);
    ss[v] = 1.0f / fmaxf(sqrtf(s), 1e-12f);
  }
  // Write ego_new in place and accumulate normalized residual.
#pragma unroll
  for (int v = 0; v < 8; ++v) {
    size_t rbase = (size_t)(row0 + v + 8 * h) * EMBED + m;
    float  sc    = ss[v];
#pragma unroll
    for (int j = 0; j < 8; ++j) {
      float x = acc[j][v];
      ego[rbase + j * 16] = x;
      out[rbase + j * 16] += x * sc;
    }
  }
}

extern "C" void kernel_launch(void* const* d_in, const int* in_sizes, int n_in,
                              void* d_out, int out_size, void* d_ws, size_t ws_size,
                              hipStream_t stream) {
  const int*   edge_row = (const int*)d_in[0];
  const int*   edge_col = (const int*)d_in[1];
  const float* edge_val = (const float*)d_in[2];
  const float* o_emb    = (const float*)d_in[3];
  const int*   u_id     = (const int*)d_in[4];
  const float* user_emb = (const float*)d_in[5];
  const float* fc1_w    = (const float*)d_in[6];
  const float* fc1_b    = (const float*)d_in[7];
  const float* fc2_w    = (const float*)d_in[8];
  const float* fc2_b    = (const float*)d_in[9];
  float*       out      = (float*)d_out;

  // Workspace layout: ego (200000x128 f32) | side (200000x128 f32) = 204.8 MB
  float* ego  = (float*)d_ws;
  float* side = ego + (size_t)N_NODES * EMBED;

  const long n4        = (long)N_NODES * (EMBED / 4);        // float4 elements
  const int  vec_grid  = (int)((n4 + 255) / 256);            // 25000
  const int  spmm_grid = (NNZ_E + 7) / 8;                    // 8 waves/block
  const int  gemm_grid = (N_NODES + 127) / 128;              // 128 rows/block

  o2u_init_kernel<<<vec_grid, 256, 0, stream>>>(user_emb, u_id, o_emb, ego, out);

  for (int l = 0; l < N_LAYERS; ++l) {
    o2u_zero_kernel<<<vec_grid, 256, 0, stream>>>((float4*)side, n4);
    o2u_spmm_kernel<<<spmm_grid, 256, 0, stream>>>(edge_row, edge_col, edge_val,
                                                   ego, side);
    o2u_layer_kernel<<<gemm_grid, 256, 0, stream>>>(
        side, ego,
        fc1_w + (size_t)l * EMBED * EMBED, fc1_b + (size_t)l * EMBED,
        fc2_w + (size_t)l * EMBED * EMBED, fc2_b + (size_t)l * EMBED,
        out);
  }
}